// LMLLoss_41841571397805
// MI455X (gfx1250) — compile-verified
//
#include <hip/hip_runtime.h>

// ---------------- problem constants (match reference) ----------------
#define CCOLS   32000            // N_CLASSES
#define C4      (CCOLS / 4)      // real float4 chunks per row (8000)
#define C4PAD   8192             // padded chunks: 8192*16B = 128 KB, 8 per thread
#define CPADF   (C4PAD * 4)      // padded floats (32768)
#define KBUDGET 5.0f             // LML budget K
#define BLOCK   1024             // 32 wave32s -> fills one WGP
#define CHUNKS  (C4PAD / BLOCK)  // exactly 8 chunks per thread (uniform!)
#define MAXIT   24               // safeguarded Newton/bisection evals

// CDNA5 async direct-to-LDS path (probe-verified on this toolchain)
#if defined(__has_builtin)
#if __has_builtin(__builtin_amdgcn_global_load_async_to_lds_b128) && \
    __has_builtin(__builtin_amdgcn_s_wait_asynccnt)
#define USE_ASYNC_LDS 1
#endif
#endif

typedef int v4i __attribute__((vector_size(4 * sizeof(int))));
typedef __attribute__((address_space(1))) v4i* gv4i_p;   // global int4*
typedef __attribute__((address_space(3))) v4i* lv4i_p;   // LDS int4*

// ---------------- wave32 / block reductions ----------------
__device__ __forceinline__ float wredsum(float v) {
#pragma unroll
  for (int o = 16; o; o >>= 1) v += __shfl_xor(v, o, 32);
  return v;
}

// simultaneous block-sum of two values; result broadcast to all threads
__device__ __forceinline__ void block_sum2(float a, float b,
                                           float* r1, float* r2, float* out2,
                                           float& oa, float& ob) {
  const int lane = threadIdx.x & 31;
  const int w    = threadIdx.x >> 5;
  a = wredsum(a); b = wredsum(b);
  if (lane == 0) { r1[w] = a; r2[w] = b; }
  __syncthreads();
  if (w == 0) {
    const int nw = (int)(blockDim.x >> 5);
    float ta = (lane < nw) ? r1[lane] : 0.0f;
    float tb = (lane < nw) ? r2[lane] : 0.0f;
    ta = wredsum(ta); tb = wredsum(tb);
    if (lane == 0) { out2[0] = ta; out2[1] = tb; }
  }
  __syncthreads();
  oa = out2[0]; ob = out2[1];
}

__device__ __forceinline__ void block_minmax(float mn, float mx,
                                             float* r1, float* r2, float* out2,
                                             float& omn, float& omx) {
  const int lane = threadIdx.x & 31;
  const int w    = threadIdx.x >> 5;
#pragma unroll
  for (int o = 16; o; o >>= 1) {
    mn = fminf(mn, __shfl_xor(mn, o, 32));
    mx = fmaxf(mx, __shfl_xor(mx, o, 32));
  }
  if (lane == 0) { r1[w] = mn; r2[w] = mx; }
  __syncthreads();
  if (w == 0) {
    const int nw = (int)(blockDim.x >> 5);
    float tn = (lane < nw) ? r1[lane] :  3.4e38f;
    float tx = (lane < nw) ? r2[lane] : -3.4e38f;
#pragma unroll
    for (int o = 16; o; o >>= 1) {
      tn = fminf(tn, __shfl_xor(tn, o, 32));
      tx = fmaxf(tx, __shfl_xor(tx, o, 32));
    }
    if (lane == 0) { out2[0] = tn; out2[1] = tx; }
  }
  __syncthreads();
  omn = out2[0]; omx = out2[1];
}

// ---------------- sigmoid-sum evaluators over the LDS-resident e_i table ----
// sigma(x_i + nu) = t/(1+t) with t = e_i * E, E = exp(nu).
// Pad entries hold e=0 -> t=0 -> s=0: contribute nothing to either sum.
// Uniform 8-iteration loops: compile-time trip count, ds loads batched.
__device__ __forceinline__ void eval_fast(const float4* s4, float E,
                                          float& sg, float& sp) {
  float g = 0.0f, p = 0.0f;
  const int tid = threadIdx.x;
#pragma unroll
  for (int k = 0; k < CHUNKS; ++k) {
    float4 e = s4[tid + k * BLOCK];
    float t0 = e.x * E, t1 = e.y * E, t2 = e.z * E, t3 = e.w * E;
    float s0 = t0 * __builtin_amdgcn_rcpf(1.0f + t0);   // v_rcp_f32 (~1 ulp)
    float s1 = t1 * __builtin_amdgcn_rcpf(1.0f + t1);
    float s2 = t2 * __builtin_amdgcn_rcpf(1.0f + t2);
    float s3 = t3 * __builtin_amdgcn_rcpf(1.0f + t3);
    g += (s0 + s1) + (s2 + s3);
    p = fmaf(s0, 1.0f - s0, p);
    p = fmaf(s1, 1.0f - s1, p);
    p = fmaf(s2, 1.0f - s2, p);
    p = fmaf(s3, 1.0f - s3, p);
  }
  sg = g; sp = p;
}

__device__ __forceinline__ void eval_precise(const float4* s4, float E,
                                             float& sg, float& sp) {
  float g = 0.0f, p = 0.0f;
  const int tid = threadIdx.x;
#pragma unroll
  for (int k = 0; k < CHUNKS; ++k) {
    float4 e = s4[tid + k * BLOCK];
    float t0 = e.x * E, t1 = e.y * E, t2 = e.z * E, t3 = e.w * E;
    float s0 = t0 / (1.0f + t0);
    float s1 = t1 / (1.0f + t1);
    float s2 = t2 / (1.0f + t2);
    float s3 = t3 / (1.0f + t3);
    g += (s0 + s1) + (s2 + s3);
    p = fmaf(s0, 1.0f - s0, p);
    p = fmaf(s1, 1.0f - s1, p);
    p = fmaf(s2, 1.0f - s2, p);
    p = fmaf(s3, 1.0f - s3, p);
  }
  sg = g; sp = p;
}

// ---------------- one workgroup per row ----------------
__global__ __launch_bounds__(BLOCK)
void lml_row_kernel(const float* __restrict__ x, const int* __restrict__ y,
                    float* __restrict__ row_loss) {
  extern __shared__ __align__(16) float s_e[];   // CPADF floats: x, then exp(xn)
  __shared__ float s_r1[32];
  __shared__ float s_r2[32];
  __shared__ float s_b[2];

  const int tid = threadIdx.x;
  const int row = blockIdx.x;
  const float* __restrict__ xrow = x + (size_t)row * CCOLS;

  // -- Stage 1: HBM -> LDS once via CDNA5 async direct-to-LDS (ASYNCcnt);
  //    zero-fill the 192 pad chunks so all later passes are uniform. --
  {
    float4* l4 = (float4*)s_e;
    if (tid < (C4PAD - C4)) {
      float4 z; z.x = 0.0f; z.y = 0.0f; z.z = 0.0f; z.w = 0.0f;
      l4[C4 + tid] = z;
    }
  }
#ifdef USE_ASYNC_LDS
  {
    gv4i_p gsrc = (gv4i_p)(void*)const_cast<float*>(xrow);
    lv4i_p ldst = (lv4i_p)(void*)s_e;
    for (int i = tid; i < C4; i += BLOCK)
      __builtin_amdgcn_global_load_async_to_lds_b128(gsrc + i, ldst + i, 0, 0);
    __builtin_amdgcn_s_wait_asynccnt(0);
  }
#else
  {
    const float4* g4 = (const float4*)xrow;
    float4* l4 = (float4*)s_e;
    for (int i = tid; i < C4; i += BLOCK) l4[i] = g4[i];
  }
#endif
  __syncthreads();

  const float4* s4c = (const float4*)s_e;

  // -- Stage 2: ||x||^2 (uniform over pad; pads are 0) --
  float ss = 0.0f;
#pragma unroll
  for (int k = 0; k < CHUNKS; ++k) {
    float4 v = s4c[tid + k * BLOCK];
    ss = fmaf(v.x, v.x, ss); ss = fmaf(v.y, v.y, ss);
    ss = fmaf(v.z, v.z, ss); ss = fmaf(v.w, v.w, ss);
  }
  float ssum, unused;
  block_sum2(ss, 0.0f, s_r1, s_r2, s_b, ssum, unused);
  (void)unused;
  const float inv = 1.0f / fmaxf(sqrtf(ssum), 1e-12f);

  // -- Stage 3: in place e_i = exp(x_i * inv) on real chunks only
  //    (pads keep 0.0 -> zero contribution later); track min/max of xn --
  float mx = -3.4e38f, mn = 3.4e38f;
  {
    float4* s4 = (float4*)s_e;
    for (int i = tid; i < C4; i += BLOCK) {
      float4 v = s4[i];
      float a0 = v.x * inv, a1 = v.y * inv, a2 = v.z * inv, a3 = v.w * inv;
      mx = fmaxf(mx, fmaxf(fmaxf(a0, a1), fmaxf(a2, a3)));
      mn = fminf(mn, fminf(fminf(a0, a1), fminf(a2, a3)));
      v.x = __expf(a0); v.y = __expf(a1); v.z = __expf(a2); v.w = __expf(a3);
      s4[i] = v;
    }
  }
  block_minmax(mn, mx, s_r1, s_r2, s_b, mn, mx);
  float lo = -mx - 20.0f;   // g(lo) < 0  (same brackets as reference)
  float hi = -mn + 20.0f;   // g(hi) > 0

  // -- Stage 4: bracket-safeguarded Newton on g(nu) = sum sigma - K.
  //    Reference's 60 f32 bisections resolve nu far below f32 ulp, i.e. to
  //    the f32-exact root; this converges to the same root in <= MAXIT evals.
  float nu = 0.5f * (lo + hi);
  float g, gp;
  for (int it = 0; it < MAXIT; ++it) {
    const float E = __expf(nu);
    float lg, lp;
    eval_fast(s4c, E, lg, lp);
    block_sum2(lg, lp, s_r1, s_r2, s_b, g, gp);   // g,gp bitwise-uniform
    g -= KBUDGET;
    if (g < 0.0f) lo = nu; else hi = nu;
    if (fabsf(g) <= 1e-4f) break;                 // uniform branch
    const float cand = nu - g / gp;               // Newton
    nu = (cand > lo && cand < hi) ? cand : 0.5f * (lo + hi);  // safeguard
  }

  // -- Stage 5: reference's final (differentiable) Newton step, precise --
  {
    const float E = __expf(nu);
    float lg, lp;
    eval_precise(s4c, E, lg, lp);
    block_sum2(lg, lp, s_r1, s_r2, s_b, g, gp);
    g -= KBUDGET;
    const float nu2 = nu - g / gp;

    if (tid == 0) {
      const int cls = y[row];
      const float xy = xrow[cls] * inv;                       // one global load
      const float py = 1.0f / (1.0f + expf(-(xy + nu2)));     // sigma(xn_y+nu)
      row_loss[row] = -logf(py + 1e-8f);
    }
  }
}

// ---------------- deterministic fixed-order mean over rows ----------------
__global__ __launch_bounds__(1024)
void mean_kernel(const float* __restrict__ rl, float* __restrict__ out, int n) {
  __shared__ float r1[32];
  __shared__ float r2[32];
  __shared__ float b2[2];
  float v = 0.0f;
  for (int i = threadIdx.x; i < n; i += 1024) v += rl[i];
  float s, d;
  block_sum2(v, 0.0f, r1, r2, b2, s, d);
  (void)d;
  if (threadIdx.x == 0) out[0] = s / (float)n;
}

// ---------------- entry point ----------------
extern "C" void kernel_launch(void* const* d_in, const int* in_sizes, int n_in,
                              void* d_out, int out_size, void* d_ws, size_t ws_size,
                              hipStream_t stream) {
  (void)n_in; (void)out_size; (void)ws_size;
  const float* x = (const float*)d_in[0];     // [B, 32000] f32
  const int*   y = (const int*)d_in[1];       // [B] int32
  float* out = (float*)d_out;                 // scalar f32
  float* rl  = (float*)d_ws;                  // per-row losses (B floats)
  const int rows = in_sizes[1];

  // one WGP-filling block per row; padded row = exactly 128 KB of 320 KB LDS
  lml_row_kernel<<<rows, BLOCK, CPADF * sizeof(float), stream>>>(x, y, rl);
  mean_kernel<<<1, 1024, 0, stream>>>(rl, out, rows);
}